// GATConv_20160576487461
// MI455X (gfx1250) — compile-verified
//
#include <hip/hip_runtime.h>
#include <hip/hip_bf16.h>
#include <math.h>

typedef __attribute__((ext_vector_type(2))) float v2f;
typedef __attribute__((ext_vector_type(8))) float v8f;

#define IN_F   256
#define OUT_F  32
#define HEADS  8
#define COLS   256   /* OUT_F * HEADS */
#define WP_ELEMS (128 * COLS * 2)   /* 64 k4-steps * 2 halves * 256 cols * 2 = 65536 floats */

// ---------------------------------------------------------------------------
// One-time repack of W[256,256] into WMMA-lane order so the GEMM hot loop can
// fetch each 4x16 B tile with a single coalesced b64 load per lane:
//   Wp[((k4*2 + half)*256 + n)*2 + j] = W[(k4*4 + 2*half + j)*256 + n]
// ---------------------------------------------------------------------------
__global__ void pack_W_kernel(const float* __restrict__ W, float* __restrict__ Wp) {
  const int idx = blockIdx.x * blockDim.x + threadIdx.x;   // 0 .. 65535
  if (idx >= WP_ELEMS) return;
  const int j    = idx & 1;
  const int n    = (idx >> 1) & (COLS - 1);
  const int p    = idx >> 9;          // k4*2 + half
  const int k4   = p >> 1;
  const int half = p & 1;
  Wp[idx] = W[(size_t)(k4 * 4 + 2 * half + j) * COLS + n];
}

// ---------------------------------------------------------------------------
// Projection GEMM: Zp[N,256] = Z[N,256] @ W[256,256] + b, full fp32 via
// V_WMMA_F32_16X16X4_F32. Block = 128 threads = 4 waves; wave w computes the
// 16x64 strip (cols w*64 .. w*64+63) of the block's 16-row tile.
// Hot loop: 1 A b64 + 4 B b64 loads per 4 WMMAs.
// ---------------------------------------------------------------------------
__global__ void __launch_bounds__(128)
proj_gemm_wmma(const float* __restrict__ Z, const float* __restrict__ Wp,
               const float* __restrict__ bias, float* __restrict__ Zp, int N) {
  const int lane = threadIdx.x & 31;
  const int wave = threadIdx.x >> 5;      // 0..3 -> 64-col strip
  const int l16  = lane & 15;
  const int half = lane >> 4;             // 0 or 1
  const int m0   = blockIdx.x << 4;       // 16-row tile
  const int n0   = wave << 6;

  // A-matrix 16x4 fp32 layout (ISA 7.12.2): lanes 0-15 -> K=k..k+1, lanes 16-31 -> K=k+2..k+3
  const int mA = m0 + l16;
  const float* zrow = Z + (size_t)(mA < N ? mA : (N - 1)) * IN_F;

  // Packed-B base for this lane: p = half at k=0; advances by 2 rows (1024 floats) per k-step
  const float* wb = Wp + ((size_t)half * COLS + (n0 + l16)) * 2;

  v8f acc0 = {}; v8f acc1 = {}; v8f acc2 = {}; v8f acc3 = {};

  #pragma unroll 4
  for (int k = 0; k < IN_F; k += 4) {
    v2f a  = *reinterpret_cast<const v2f*>(zrow + k + 2 * half);  // 8B-aligned
    v2f b0 = *reinterpret_cast<const v2f*>(wb);
    v2f b1 = *reinterpret_cast<const v2f*>(wb + 32);   // n += 16
    v2f b2 = *reinterpret_cast<const v2f*>(wb + 64);   // n += 32
    v2f b3 = *reinterpret_cast<const v2f*>(wb + 96);   // n += 48
    wb += 2 * COLS * 2;

    acc0 = __builtin_amdgcn_wmma_f32_16x16x4_f32(false, a, false, b0, (short)0, acc0, false, false);
    acc1 = __builtin_amdgcn_wmma_f32_16x16x4_f32(false, a, false, b1, (short)0, acc1, false, false);
    acc2 = __builtin_amdgcn_wmma_f32_16x16x4_f32(false, a, false, b2, (short)0, acc2, false, false);
    acc3 = __builtin_amdgcn_wmma_f32_16x16x4_f32(false, a, false, b3, (short)0, acc3, false, false);
  }

  // D layout: VGPR v, lanes 0-15 -> row m0+v, lanes 16-31 -> row m0+v+8; col = n + l16
  v8f accs[4] = {acc0, acc1, acc2, acc3};
  #pragma unroll
  for (int t = 0; t < 4; ++t) {
    const int n  = n0 + 16 * t + l16;
    const float bv = bias[n];
    #pragma unroll
    for (int v = 0; v < 8; ++v) {
      const int mrow = m0 + v + 8 * half;
      if (mrow < N) Zp[(size_t)mrow * COLS + n] = accs[t][v] + bv;
    }
  }
}

// ---------------------------------------------------------------------------
// Per-node attention logits e_l/e_r + init of segment max (-inf) / sum (0).
// One thread per (node, head).
// ---------------------------------------------------------------------------
__global__ void attn_logits_init(const float* __restrict__ Zp,
                                 const float* __restrict__ a_l, const float* __restrict__ a_r,
                                 float* __restrict__ e_l, float* __restrict__ e_r,
                                 float* __restrict__ mbuf, float* __restrict__ sbuf, int NH) {
  const int idx = blockIdx.x * blockDim.x + threadIdx.x;
  if (idx >= NH) return;
  const int n = idx >> 3;
  const int h = idx & 7;
  const float* z = Zp + (size_t)n * COLS + h;
  float al = 0.f, ar = 0.f;
  #pragma unroll
  for (int o = 0; o < OUT_F; ++o) {
    const float zv = z[o * HEADS];
    al = fmaf(zv, a_l[o * HEADS + h], al);
    ar = fmaf(zv, a_r[o * HEADS + h], ar);
  }
  e_l[idx]  = al;
  e_r[idx]  = ar;
  mbuf[idx] = -INFINITY;
  sbuf[idx] = 0.f;
}

__device__ __forceinline__ float leaky01(float v) {
  return v > 0.f ? v : 0.01f * v;
}

// Float atomic max via sign-split integer atomics (correct for mixed signs, -inf init).
__device__ __forceinline__ void atomicMaxFloat(float* addr, float val) {
  if (val >= 0.f) atomicMax((int*)addr, __float_as_int(val));
  else            atomicMin((unsigned int*)addr, __float_as_uint(val));
}

// ---------------------------------------------------------------------------
// Segment max over edges: one thread per (edge, head).
// ---------------------------------------------------------------------------
__global__ void edge_max_kernel(const int* __restrict__ row, const int* __restrict__ col,
                                const float* __restrict__ e_l, const float* __restrict__ e_r,
                                float* __restrict__ mbuf, int EH) {
  const int idx = blockIdx.x * blockDim.x + threadIdx.x;
  if (idx >= EH) return;
  const int e = idx >> 3, h = idx & 7;
  const int r = row[e], c = col[e];
  const float v = leaky01(e_l[r * HEADS + h] + e_r[c * HEADS + h]);
  atomicMaxFloat(&mbuf[r * HEADS + h], v);
}

// ---------------------------------------------------------------------------
// Segment sum of exp(e - m): one thread per (edge, head).
// ---------------------------------------------------------------------------
__global__ void edge_expsum_kernel(const int* __restrict__ row, const int* __restrict__ col,
                                   const float* __restrict__ e_l, const float* __restrict__ e_r,
                                   const float* __restrict__ mbuf, float* __restrict__ sbuf, int EH) {
  const int idx = blockIdx.x * blockDim.x + threadIdx.x;
  if (idx >= EH) return;
  const int e = idx >> 3, h = idx & 7;
  const int r = row[e], c = col[e];
  const float v  = leaky01(e_l[r * HEADS + h] + e_r[c * HEADS + h]);
  const float ex = __expf(v - mbuf[r * HEADS + h]);
  atomicAdd(&sbuf[r * HEADS + h], ex);
}

// ---------------------------------------------------------------------------
// Weighted message scatter: one 256-thread block per edge; thread c handles
// column c = o*H + h. Coalesced gather of Zp[col], coalesced atomic add into
// out[row]. Both live in L2 (51 MB each << 192 MB).
// ---------------------------------------------------------------------------
__global__ void __launch_bounds__(256)
edge_scatter_kernel(const int* __restrict__ row, const int* __restrict__ col,
                    const float* __restrict__ e_l, const float* __restrict__ e_r,
                    const float* __restrict__ mbuf, const float* __restrict__ sbuf,
                    const float* __restrict__ Zp, float* __restrict__ out) {
  const int e = blockIdx.x;
  const int c = threadIdx.x;           // 0..255
  const int r  = row[e];
  const int cl = col[e];
  const int h  = c & 7;
  const float v    = leaky01(e_l[r * HEADS + h] + e_r[cl * HEADS + h]);
  const float attn = __expf(v - mbuf[r * HEADS + h]) / sbuf[r * HEADS + h];
  atomicAdd(&out[(size_t)r * COLS + c], attn * Zp[(size_t)cl * COLS + c]);
}

extern "C" void kernel_launch(void* const* d_in, const int* in_sizes, int n_in,
                              void* d_out, int out_size, void* d_ws, size_t ws_size,
                              hipStream_t stream) {
  const float* Z   = (const float*)d_in[0];
  const int*   row = (const int*)d_in[1];
  const int*   col = (const int*)d_in[2];
  const float* W   = (const float*)d_in[3];
  const float* bv  = (const float*)d_in[4];
  const float* a_l = (const float*)d_in[5];
  const float* a_r = (const float*)d_in[6];
  float* out = (float*)d_out;

  const int N = in_sizes[0] / IN_F;
  const int E = in_sizes[1];

  // Workspace layout
  float* Zp  = (float*)d_ws;                      // N*256
  float* e_l = Zp  + (size_t)N * COLS;            // N*8
  float* e_r = e_l + (size_t)N * HEADS;           // N*8
  float* mb  = e_r + (size_t)N * HEADS;           // N*8
  float* sb  = mb  + (size_t)N * HEADS;           // N*8
  float* Wp  = sb  + (size_t)N * HEADS;           // 65536 (256 KB packed W)

  hipMemsetAsync(out, 0, (size_t)out_size * sizeof(float), stream);

  pack_W_kernel<<<(WP_ELEMS + 255) / 256, 256, 0, stream>>>(W, Wp);

  proj_gemm_wmma<<<dim3((N + 15) / 16), 128, 0, stream>>>(Z, Wp, bv, Zp, N);

  const int NH = N * HEADS;
  attn_logits_init<<<(NH + 255) / 256, 256, 0, stream>>>(Zp, a_l, a_r, e_l, e_r, mb, sb, NH);

  const int EH = E * HEADS;
  edge_max_kernel   <<<(EH + 255) / 256, 256, 0, stream>>>(row, col, e_l, e_r, mb, EH);
  edge_expsum_kernel<<<(EH + 255) / 256, 256, 0, stream>>>(row, col, e_l, e_r, mb, sb, EH);
  edge_scatter_kernel<<<E, 256, 0, stream>>>(row, col, e_l, e_r, mb, sb, Zp, out);
}